// MSDeformLineAttn_48533130445137
// MI455X (gfx1250) — compile-verified
//
#include <hip/hip_runtime.h>
#include <hip/hip_bf16.h>
#include <math.h>

// ---------------------------------------------------------------------------
// MS deformable line attention for MI455X (gfx1250, wave32, WMMA + TDM).
//
// Stage 1 (proj_loc): fused GEMM  query(8192x256) x [w_ratio|w_attn](256x256)
//   via V_WMMA_F32_16X16X4_F32 (exact fp32), then per-head softmax over P=16
//   and sampling-location math -> float4{x,y,attnW,0} per (q,h,p) in d_ws.
// Stage 2 (transpose): value (NM,D,HW) -> valueT (NM,HW,D) staged through LDS
//   using the Tensor Data Mover (tensor_load_to_lds, TENSORcnt) so every
//   bilinear corner fetch in stage 3 becomes ONE 128B transaction instead of
//   32 scattered 4B-per-line transactions (values are L2-resident: 178MB of
//   192MB L2; transpose costs ~2x178MB HBM ~= 15us @ 23.3 TB/s, saves ~30x
//   L2 gather traffic). Falls back to direct gather if ws_size is too small.
// Stage 3 (sample): one wave32 per (query,head), lane = channel d in [0,32),
//   16-point bilinear gather + attention-weighted accumulate.
// ---------------------------------------------------------------------------

typedef __attribute__((ext_vector_type(2))) float v2f;
typedef __attribute__((ext_vector_type(8))) float v8f;
typedef __attribute__((ext_vector_type(4))) unsigned int v4u;
typedef __attribute__((ext_vector_type(8))) int v8i;
typedef __attribute__((ext_vector_type(4))) int v4i;

#define QS_STRIDE 258   // 16-row query tile in LDS, padded (banks shift 2/row)

// ============================ Stage 1: GEMM + loc ==========================

__global__ __launch_bounds__(256)
void msdla_proj_loc_kernel(const float* __restrict__ query,
                           const float* __restrict__ refp,
                           const float* __restrict__ w_ratio,
                           const float* __restrict__ b_ratio,
                           const float* __restrict__ w_attn,
                           const float* __restrict__ b_attn,
                           float4* __restrict__ ws)
{
    __shared__ float qs[16 * QS_STRIDE];  // query tile   16 x 256 (padded)
    __shared__ float os[16 * 256];        // gemm output  16 x (128 ratio|128 attn)

    const int tid     = threadIdx.x;
    const int rowBase = blockIdx.x * 16;          // 512 blocks over 8192 rows

    {   // load 16x256 query tile into LDS (float4 vectorized)
        const float4* q4 = reinterpret_cast<const float4*>(query + (size_t)rowBase * 256);
        for (int i = tid; i < 16 * 64; i += 256) {
            const int r  = i >> 6;
            const int c4 = i & 63;
            float4 v = q4[(size_t)r * 64 + c4];
            float* dst = &qs[r * QS_STRIDE + c4 * 4];
            dst[0] = v.x; dst[1] = v.y; dst[2] = v.z; dst[3] = v.w;
        }
    }
    __syncthreads();

    // WMMA GEMM: each of 8 waves owns two 16-col tiles of the 256 cols
    const int lane = tid & 31;
    const int wave = tid >> 5;
    const int hi   = lane >> 4;       // half-wave selector (ISA fp32 layouts)
    const int arow = lane & 15;

    const int gc0 = (wave * 2 + 0) * 16 + arow;
    const int gc1 = (wave * 2 + 1) * 16 + arow;
    const float* W0 = (gc0 < 128) ? w_ratio : w_attn;
    const float* W1 = (gc1 < 128) ? w_ratio : w_attn;
    const int c0 = gc0 & 127;
    const int c1 = gc1 & 127;

    v8f acc0 = {};
    v8f acc1 = {};

#pragma unroll 4
    for (int k = 0; k < 256; k += 4) {
        const int kk = k + 2 * hi;    // lo lanes: K=k,k+1   hi lanes: K=k+2,k+3
        v2f a;
        a.x = qs[arow * QS_STRIDE + kk];
        a.y = qs[arow * QS_STRIDE + kk + 1];
        v2f b0;
        b0.x = W0[(size_t)kk * 128 + c0];
        b0.y = W0[(size_t)(kk + 1) * 128 + c0];
        v2f b1;
        b1.x = W1[(size_t)kk * 128 + c1];
        b1.y = W1[(size_t)(kk + 1) * 128 + c1];
        acc0 = __builtin_amdgcn_wmma_f32_16x16x4_f32(false, a, false, b0,
                                                     (short)0, acc0, false, false);
        acc1 = __builtin_amdgcn_wmma_f32_16x16x4_f32(false, a, false, b1,
                                                     (short)0, acc1, false, false);
    }

    const float bias0 = (gc0 < 128) ? b_ratio[c0] : b_attn[c0];
    const float bias1 = (gc1 < 128) ? b_ratio[c1] : b_attn[c1];
#pragma unroll
    for (int r = 0; r < 8; ++r) {
        const int orow = r + 8 * hi;  // C/D layout: VGPR r -> row r + 8*hi
        os[orow * 256 + gc0] = acc0[r] + bias0;
        os[orow * 256 + gc1] = acc1[r] + bias1;
    }
    __syncthreads();

    // softmax over P=16 + sampling-location math: 1 thread per (q,h)
    if (tid < 128) {
        const int qr   = tid >> 3;
        const int h    = tid & 7;
        const int grow = rowBase + qr;

        const float4 rp = reinterpret_cast<const float4*>(refp)[grow];
        const float vx = rp.x - rp.z;
        const float vy = rp.y - rp.w;
        const float cx = 0.5f * (rp.x + rp.z);
        const float cy = 0.5f * (rp.y + rp.w);

        float lg[16];
        float mx = -1e30f;
#pragma unroll
        for (int p = 0; p < 16; ++p) {
            lg[p] = os[qr * 256 + 128 + h * 16 + p];
            mx = fmaxf(mx, lg[p]);
        }
        float s = 0.f;
#pragma unroll
        for (int p = 0; p < 16; ++p) {
            lg[p] = __expf(lg[p] - mx);
            s += lg[p];
        }
        const float inv = 1.f / s;

        const float dims[4] = {128.f, 64.f, 32.f, 16.f};
        float4* o4 = ws + ((size_t)grow * 128 + h * 16);
#pragma unroll
        for (int p = 0; p < 16; ++p) {
            const float ratio = os[qr * 256 + h * 16 + p];
            const float lx = cx + ratio * 0.125f * vx;  // scale=1/4, *vec*0.5
            const float ly = cy + ratio * 0.125f * vy;
            const float d  = dims[p >> 2];
            float4 o;
            o.x = lx * d - 0.5f;
            o.y = ly * d - 0.5f;
            o.z = lg[p] * inv;
            o.w = 0.f;
            o4[p] = o;
        }
    }
}

// ====================== Stage 2: TDM-staged transpose ======================
// One block per 32(d) x 64(s) tile. TDM pulls the strided tile into LDS with
// a 1-dword pad every 64 dwords (row stride 65 -> conflict-free transposed
// reads), then threads write (s,d)-major, fully coalesced, to valueT.

__global__ __launch_bounds__(256)
void msdla_transpose_kernel(const float* __restrict__ src,
                            float* __restrict__ dst,
                            int HW)
{
    __shared__ float tile[32 * 65 + 32];   // slack for trailing TDM pad

    const int chunks = HW >> 6;                    // 64-wide spatial chunks
    const int nm     = blockIdx.x / chunks;
    const int sBase  = (blockIdx.x % chunks) << 6;
    const float* gsrc = src + (size_t)nm * 32 * HW + sBase;

#if __has_builtin(__builtin_amdgcn_tensor_load_to_lds)
    if (threadIdx.x < 32) {   // TDM issues per-wave (EXEC ignored): wave 0 only
        const unsigned long long ga = (unsigned long long)(uintptr_t)gsrc;
        v4u g0;
        g0.x = 1u;                                  // count=1, gather off
        g0.y = 0u;                                  // lds_addr = 0 (tile base)
        g0.z = (unsigned)(ga & 0xFFFFFFFFu);        // global_addr[31:0]
        g0.w = (unsigned)((ga >> 32) & 0x01FFFFFFu) // global_addr[56:32]
               | 0x80000000u;                       // type=2 (image)
        v8i g1;
        g1[0] = (int)((2u << 16)     // data_size = 4B
                    | (1u << 20)     // pad_enable
                    | (5u << 22));   // pad_interval: 64 dwords; pad_amount=1dw
        g1[1] = (int)(((unsigned)HW & 0xFFFFu) << 16);        // tensor_dim0 lo
        g1[2] = (int)(((unsigned)HW >> 16) | (32u << 16));    // dim0 hi | dim1
        g1[3] = (int)(64u << 16);                             // tile_dim0 = 64
        g1[4] = 32;                                           // tile_dim1 = 32
        g1[5] = HW;                                 // tensor_dim0_stride[31:0]
        g1[6] = 0;
        g1[7] = 0;
        v4i gz4 = {0, 0, 0, 0};
        v8i gz8 = {0, 0, 0, 0, 0, 0, 0, 0};
        // clang-23 / therock flavor: 6 args, 5th operand is int32x8
        __builtin_amdgcn_tensor_load_to_lds(g0, g1, gz4, gz4, gz8, 0);
#if __has_builtin(__builtin_amdgcn_s_wait_tensorcnt)
        __builtin_amdgcn_s_wait_tensorcnt(0);
#else
        asm volatile("s_wait_tensorcnt 0x0" ::: "memory");
#endif
    }
#else
    // Fallback: coalesced VMEM loads + LDS stores
    for (int i = threadIdx.x; i < 32 * 64; i += 256) {
        const int d = i >> 6;
        const int s = i & 63;
        tile[d * 65 + s] = gsrc[(size_t)d * HW + s];
    }
#endif
    __syncthreads();

    // write transposed: consecutive lanes -> consecutive d -> coalesced
    for (int o = threadIdx.x; o < 32 * 64; o += 256) {
        const int s = o >> 5;
        const int d = o & 31;
        dst[((size_t)nm * HW + sBase + s) * 32 + d] = tile[d * 65 + s];
    }
}

// ========================= Stage 3: bilinear sample ========================

template <bool TRANS>
__global__ __launch_bounds__(256)
void msdla_sample_kernel(const float4* __restrict__ ws,
                         const float* __restrict__ v0,
                         const float* __restrict__ v1,
                         const float* __restrict__ v2,
                         const float* __restrict__ v3,
                         float* __restrict__ out)
{
    const int tid  = threadIdx.x;
    const int lane = tid & 31;                        // lane == channel d
    const int wid  = blockIdx.x * 8 + (tid >> 5);     // 65536 (q,h) waves
    const int m    = wid & 7;
    const int row  = wid >> 3;                        // n*1024+q in [0,8192)
    const int n    = row >> 10;
    const int nm   = n * 8 + m;

    const float4* p4 = ws + ((size_t)row * 128 + m * 16);
    float acc = 0.f;

#pragma unroll
    for (int p = 0; p < 16; ++p) {
        const float4 s = p4[p];                 // broadcast load
        const int lvl = p >> 2;
        const int Wl  = 128 >> lvl;             // 128,64,32,16 (square)
        const int HW  = Wl * Wl;
        const float* vlvl;
        if      (lvl == 0) vlvl = v0;
        else if (lvl == 1) vlvl = v1;
        else if (lvl == 2) vlvl = v2;
        else               vlvl = v3;
        // TRANS: (NM,HW,D) -> corner = 128B contiguous; else (NM,D,HW)
        const float* vbase = TRANS
            ? vlvl + (size_t)nm * HW * 32
            : vlvl + ((size_t)nm * 32 + lane) * HW;

        const float x   = s.x, y = s.y, aw = s.z;
        const float x0f = floorf(x);
        const float y0f = floorf(y);
        const int   x0  = (int)x0f;
        const int   y0  = (int)y0f;
        const float wx  = x - x0f;
        const float wy  = y - y0f;

        const bool xin0 = (x0 >= 0)     && (x0 < Wl);
        const bool xin1 = (x0 + 1 >= 0) && (x0 + 1 < Wl);
        const bool yin0 = (y0 >= 0)     && (y0 < Wl);
        const bool yin1 = (y0 + 1 >= 0) && (y0 + 1 < Wl);

        float c00 = 0.f, c01 = 0.f, c10 = 0.f, c11 = 0.f;
        if (yin0) {
            const int base = y0 * Wl;
            if (xin0) c00 = TRANS ? vbase[(size_t)(base + x0) * 32 + lane]
                                  : vbase[base + x0];
            if (xin1) c01 = TRANS ? vbase[(size_t)(base + x0 + 1) * 32 + lane]
                                  : vbase[base + x0 + 1];
        }
        if (yin1) {
            const int base = (y0 + 1) * Wl;
            if (xin0) c10 = TRANS ? vbase[(size_t)(base + x0) * 32 + lane]
                                  : vbase[base + x0];
            if (xin1) c11 = TRANS ? vbase[(size_t)(base + x0 + 1) * 32 + lane]
                                  : vbase[base + x0 + 1];
        }
        const float bil = c00 * (1.f - wx) * (1.f - wy)
                        + c01 * wx         * (1.f - wy)
                        + c10 * (1.f - wx) * wy
                        + c11 * wx         * wy;
        acc = fmaf(aw, bil, acc);
    }

    out[(size_t)row * 256 + m * 32 + lane] = acc;   // out[n][q][m*32+d]
}

// ================================ Launch ===================================

extern "C" void kernel_launch(void* const* d_in, const int* in_sizes, int n_in,
                              void* d_out, int out_size, void* d_ws, size_t ws_size,
                              hipStream_t stream)
{
    const float* query  = (const float*)d_in[0];
    const float* refp   = (const float*)d_in[1];
    const float* wratio = (const float*)d_in[2];
    const float* bratio = (const float*)d_in[3];
    const float* wattn  = (const float*)d_in[4];
    const float* battn  = (const float*)d_in[5];
    const float* v0     = (const float*)d_in[6];
    const float* v1     = (const float*)d_in[7];
    const float* v2     = (const float*)d_in[8];
    const float* v3     = (const float*)d_in[9];
    float4* ws4         = (float4*)d_ws;
    float*  out         = (float*)d_out;

    // workspace budget: 16MB sample params + 178.25MB transposed values
    const size_t PARAM_BYTES = (size_t)8192 * 128 * 16;          // 16777216
    const size_t T0 = (size_t)64 * 16384 * 32;                   // floats
    const size_t T1 = (size_t)64 * 4096  * 32;
    const size_t T2 = (size_t)64 * 1024  * 32;
    const size_t T3 = (size_t)64 * 256   * 32;
    const size_t NEED = PARAM_BYTES + (T0 + T1 + T2 + T3) * sizeof(float);
    const bool useT = (ws_size >= NEED);

    msdla_proj_loc_kernel<<<512, 256, 0, stream>>>(query, refp, wratio, bratio,
                                                   wattn, battn, ws4);

    if (useT) {
        float* vT  = (float*)((char*)d_ws + PARAM_BYTES);
        float* t0  = vT;
        float* t1  = t0 + T0;
        float* t2  = t1 + T1;
        float* t3  = t2 + T2;
        msdla_transpose_kernel<<<16384, 256, 0, stream>>>(v0, t0, 16384);
        msdla_transpose_kernel<<< 4096, 256, 0, stream>>>(v1, t1, 4096);
        msdla_transpose_kernel<<< 1024, 256, 0, stream>>>(v2, t2, 1024);
        msdla_transpose_kernel<<<  256, 256, 0, stream>>>(v3, t3, 256);
        msdla_sample_kernel<true><<<8192, 256, 0, stream>>>(ws4, t0, t1, t2, t3, out);
    } else {
        msdla_sample_kernel<false><<<8192, 256, 0, stream>>>(ws4, v0, v1, v2, v3, out);
    }
}